// ProtoSupConLoss_41549513622029
// MI455X (gfx1250) — compile-verified
//
#include <hip/hip_runtime.h>
#include <hip/hip_bf16.h>

typedef __attribute__((ext_vector_type(2))) float v2f;
typedef __attribute__((ext_vector_type(8))) float v8f;

#define NB 8
#define NS 2048
#define ND 128
#define NC 64

__device__ __constant__ float kDimNorm = 11.313708498984760f;   // sqrt(128)
__device__ __constant__ float kInvT    = 0.126269376218010f;    // 1/(0.7*sqrt(128))

// ---------------------------------------------------------------------------
// Kernel 1: per-anchor prep — sq, rank, counts, centroid pos/neg terms.
// grid = B*(S/256) = 64 blocks, 256 threads. One thread per (b,i).
// ---------------------------------------------------------------------------
__global__ __launch_bounds__(256) void prep_kernel(
    const float* __restrict__ F, const int* __restrict__ L,
    const float* __restrict__ Cen, const float* __restrict__ Phi,
    float* __restrict__ sqOut, int* __restrict__ rankOut, int* __restrict__ cntOut,
    float* __restrict__ posCOut, float* __restrict__ negCOut)
{
    __shared__ float sCent[NC * ND];   // 32 KB
    __shared__ float sCsq[NC];
    __shared__ float sRphi[NC];
    __shared__ int   sLab[NS];         // 8 KB

    const int tid = threadIdx.x;
    const int b   = blockIdx.x >> 3;
    const int i   = ((blockIdx.x & 7) << 8) + tid;   // row within batch [0,2048)

    for (int idx = tid; idx < NC * ND; idx += 256) sCent[idx] = Cen[idx];
    for (int idx = tid; idx < NS; idx += 256)      sLab[idx]  = L[b * NS + idx];
    __syncthreads();
    if (tid < NC) {
        float s = 0.f;
        #pragma unroll 8
        for (int d = 0; d < ND; ++d) { float c = sCent[tid * ND + d]; s += c * c; }
        sCsq[tid]  = s;
        sRphi[tid] = 1.0f / (Phi[tid] * kDimNorm);
    }
    __syncthreads();

    // load my feature row (128 floats) into registers
    const float4* fr = (const float4*)(F + ((size_t)(b * NS + i)) * ND);
    float4 fv[ND / 4];
    float sqi = 0.f;
    #pragma unroll
    for (int k = 0; k < ND / 4; ++k) {
        float4 v = fr[k];
        fv[k] = v;
        sqi += v.x * v.x + v.y * v.y + v.z * v.z + v.w * v.w;
    }

    const int myLab = sLab[i];

    // centroid terms
    float tot = 0.f, pc = 0.f;
    for (int c = 0; c < NC; ++c) {
        const float4* cr = (const float4*)&sCent[c * ND];
        float dot = 0.f;
        #pragma unroll
        for (int k = 0; k < ND / 4; ++k) {
            float4 cv = cr[k];
            dot += fv[k].x * cv.x + fv[k].y * cv.y + fv[k].z * cv.z + fv[k].w * cv.w;
        }
        float cd2 = sqi + sCsq[c] - 2.0f * dot;
        float ce  = __expf(cd2 * sRphi[c]);
        tot += ce;
        if (c == myLab) pc = ce;
    }

    // rank (same-label strictly before i) and counts (all same-label incl self)
    int cnt = 0, rk = 0;
    for (int j = 0; j < NS; ++j) {
        int eq = (sLab[j] == myLab);
        cnt += eq;
        rk  += (eq && (j < i)) ? 1 : 0;
    }

    const size_t g = (size_t)b * NS + i;
    sqOut[g]   = sqi;
    rankOut[g] = rk;
    cntOut[g]  = cnt;
    posCOut[g] = pc;
    negCOut[g] = tot - pc;
}

// ---------------------------------------------------------------------------
// Kernel 2: WMMA Gram + fused exp + per-row reductions -> per-anchor term.
// grid = B*(S/16) = 1024 blocks, 256 threads (8 waves).
// Each block handles one 16-row i-tile; waves split the 128 j-tiles.
// ---------------------------------------------------------------------------
__global__ __launch_bounds__(256) void gram_kernel(
    const float* __restrict__ F, const int* __restrict__ L,
    const float* __restrict__ sqA, const int* __restrict__ rankA,
    const int* __restrict__ cntA, const float* __restrict__ posC,
    const float* __restrict__ negC, float* __restrict__ termOut)
{
    __shared__ float sRS[8][16];
    __shared__ float sPS[8][16];
    __shared__ float sER[8][16];

    const int tid  = threadIdx.x;
    const int wave = tid >> 5;
    const int lane = tid & 31;
    const int b    = blockIdx.x >> 7;
    const int i0   = (blockIdx.x & 127) << 4;

    const float* Fb  = F + (size_t)b * NS * ND;
    const int rowl = lane & 15;        // M (for A) / N (for B) within tile
    const int mh   = lane >> 4;        // lane-half selects K {0,1} vs {2,3}
    const int koff = mh << 1;

    // A operand: A[m][k] = F[i0+m][k]; lane holds row (lane&15),
    // vgpr pair covers K = 4*kk + koff, 4*kk + koff + 1  (ISA 16x4 f32 layout)
    v2f a[32];
    {
        const float* ap = Fb + (size_t)(i0 + rowl) * ND + koff;
        #pragma unroll
        for (int kk = 0; kk < 32; ++kk) a[kk] = *(const v2f*)(ap + 4 * kk);
    }

    // per-lane row constants for the 8 rows this lane-half owns (m = mh*8+r)
    float sqi[8]; int labi[8], rki[8];
    #pragma unroll
    for (int r = 0; r < 8; ++r) {
        int gi = b * NS + i0 + mh * 8 + r;
        sqi[r]  = sqA[gi];
        labi[r] = L[gi];
        rki[r]  = rankA[gi];
    }

    float rs[8], ps[8], es[8];
    #pragma unroll
    for (int r = 0; r < 8; ++r) { rs[r] = 0.f; ps[r] = 0.f; es[r] = 0.f; }

    for (int jt = wave; jt < NS / 16; jt += 8) {
        const int j0 = jt << 4;
        // B operand: B[k][n] = F[j0+n][k] — same addressing pattern as A
        const float* bp = Fb + (size_t)(j0 + rowl) * ND + koff;
        v2f breg[32];
        #pragma unroll
        for (int kk = 0; kk < 32; ++kk) breg[kk] = *(const v2f*)(bp + 4 * kk);

        v8f acc = {0.f, 0.f, 0.f, 0.f, 0.f, 0.f, 0.f, 0.f};
        #pragma unroll
        for (int kk = 0; kk < 32; ++kk) {
            acc = __builtin_amdgcn_wmma_f32_16x16x4_f32(
                false, a[kk], false, breg[kk], (short)0, acc, false, false);
        }

        const int jn   = j0 + rowl;                 // this lane's column j
        const float sqj = sqA[b * NS + jn];
        const int  labj = L[b * NS + jn];

        #pragma unroll
        for (int r = 0; r < 8; ++r) {
            // D layout: vgpr r holds row m = mh*8 + r, lane (lane&15) = col n
            float d2 = sqi[r] + sqj - 2.0f * acc[r];
            float e  = __expf(d2 * kInvT);
            rs[r] += e;
            const int im   = i0 + mh * 8 + r;
            const bool same = (labj == labi[r]);
            const bool self = (jn == im);
            ps[r] += (same && !self) ? e : 0.0f;
            es[r] += (jn == rki[r]) ? e : 0.0f;
        }
    }

    // reduce across the 16 column-lanes of each half (masks 1..8 stay in-group)
    #pragma unroll
    for (int r = 0; r < 8; ++r) {
        float a1 = rs[r], a2 = ps[r], a3 = es[r];
        #pragma unroll
        for (int msk = 8; msk >= 1; msk >>= 1) {
            a1 += __shfl_xor(a1, msk);
            a2 += __shfl_xor(a2, msk);
            a3 += __shfl_xor(a3, msk);
        }
        if (rowl == 0) {
            int mrow = mh * 8 + r;
            sRS[wave][mrow] = a1;
            sPS[wave][mrow] = a2;
            sER[wave][mrow] = a3;
        }
    }
    __syncthreads();

    if (tid < 16) {
        float R = 0.f, P = 0.f, Er = 0.f;
        #pragma unroll
        for (int w = 0; w < 8; ++w) { R += sRS[w][tid]; P += sPS[w][tid]; Er += sER[w][tid]; }
        const int gi = b * NS + i0 + tid;
        float posS = P + posC[gi];
        float negS = (R - Er) + negC[gi];
        float den  = (float)(cntA[gi] + 1);
        termOut[gi] = -__logf(posS / (den * negS));
    }
}

// ---------------------------------------------------------------------------
// Kernel 3: deterministic final reduction of B*S terms, scale by 1/S.
// ---------------------------------------------------------------------------
__global__ __launch_bounds__(256) void reduce_kernel(
    const float* __restrict__ term, float* __restrict__ out)
{
    __shared__ float sbuf[256];
    const int tid = threadIdx.x;
    float s = 0.f;
    #pragma unroll 4
    for (int k = 0; k < (NB * NS) / 256; ++k) s += term[tid * ((NB * NS) / 256) + k];
    sbuf[tid] = s;
    __syncthreads();
    for (int st = 128; st >= 1; st >>= 1) {
        if (tid < st) sbuf[tid] += sbuf[tid + st];
        __syncthreads();
    }
    if (tid == 0) out[0] = sbuf[0] * (1.0f / (float)NS);
}

// ---------------------------------------------------------------------------
extern "C" void kernel_launch(void* const* d_in, const int* in_sizes, int n_in,
                              void* d_out, int out_size, void* d_ws, size_t ws_size,
                              hipStream_t stream) {
    const float* F   = (const float*)d_in[0];   // [B,S,D] f32
    const int*   Lb  = (const int*)d_in[1];     // [B,S] int
    const float* Cen = (const float*)d_in[2];   // [C,D] f32
    const float* Phi = (const float*)d_in[3];   // [C] f32
    float* out = (float*)d_out;

    char* ws = (char*)d_ws;
    const size_t n = (size_t)NB * NS;           // 16384
    float* sq    = (float*)(ws + 0 * n * 4);
    int*   rank  = (int*)  (ws + 1 * n * 4);
    int*   cnt   = (int*)  (ws + 2 * n * 4);
    float* posC  = (float*)(ws + 3 * n * 4);
    float* negC  = (float*)(ws + 4 * n * 4);
    float* term  = (float*)(ws + 5 * n * 4);

    prep_kernel<<<NB * (NS / 256), 256, 0, stream>>>(F, Lb, Cen, Phi,
                                                     sq, rank, cnt, posC, negC);
    gram_kernel<<<NB * (NS / 16), 256, 0, stream>>>(F, Lb, sq, rank, cnt,
                                                    posC, negC, term);
    reduce_kernel<<<1, 256, 0, stream>>>(term, out);
}